// dissected_Conv2d_68968584839395
// MI455X (gfx1250) — compile-verified
//
#include <hip/hip_runtime.h>

typedef __attribute__((ext_vector_type(2))) float v2f;
typedef __attribute__((ext_vector_type(8))) float v8f;

#define IN_CH   64
#define OUT_CH  128
#define HH      56
#define WWID    56
#define NPIX    (HH * WWID)     // 3136
#define NB      4
#define NTILE   64              // spatial positions per block
#define KC      16              // input-channel chunk per LDS stage
#define APITCH  18              // floats per A row  (16 + 2 pad, even => aligned b64 pairs)
#define BPITCH  160             // floats per B kq-row ((64+16)*2 ; 160 % 64banks == 32)
#define NCHUNK  36              // 9 taps * 4 channel-chunks

__global__ __launch_bounds__(256) void conv3x3_wmma_f32(
    const float* __restrict__ x,     // [B, IN_CH, H, W]
    const float* __restrict__ Wt,    // [OUT_CH, IN_CH, 3, 3]
    const float* __restrict__ bias,  // [OUT_CH]
    float* __restrict__ out)         // [B, OUT_CH, H, W]
{
    __shared__ __align__(16) float ALds[OUT_CH * APITCH];      // [o][k]     9216 B
    __shared__ __align__(16) float BLds[(KC / 2) * BPITCH];    // [kq][n][2] 5120 B

    const int tid   = threadIdx.x;
    const int wv    = tid >> 5;
    const int lane  = tid & 31;
    const int l15   = lane & 15;
    const int lhalf = lane >> 4;        // 0: K {0,1}, 1: K {2,3}  (ISA 16x4 f32 layout)
    const int kqoff = lhalf << 1;

    const int b     = blockIdx.y;
    const int n0    = blockIdx.x * NTILE;
    const int obase = wv * 16;

    // ---------- loop-invariant per-thread staging offsets ----------
    int aOffG[8], aOffL[8];             // A: 2048 elems, 8 per thread (kk-fast, L2-friendly)
    #pragma unroll
    for (int j = 0; j < 8; ++j) {
        int idx = tid + j * 256;
        int o   = idx >> 4;
        int kk  = idx & 15;
        aOffG[j] = (o * IN_CH + kk) * 9;            // uniform (i0*9 + r) added via SGPR base
        aOffL[j] = o * APITCH + kk;
    }
    int hB[4], wB[4], xOffV[4], bOffL[4];  // B: 1024 elems, 4 per thread
    #pragma unroll
    for (int j = 0; j < 4; ++j) {
        int idx = tid + j * 256;
        int kk  = idx >> 6;
        int nn  = idx & 63;
        int n   = n0 + nn;
        int h   = n / WWID;
        int w   = n - h * WWID;
        hB[j]   = h;
        wB[j]   = w;
        xOffV[j] = (b * IN_CH + kk) * NPIX + h * WWID + w;   // uniform part added via SGPR base
        bOffL[j] = (kk >> 1) * BPITCH + nn * 2 + (kk & 1);
    }

    float aReg[8], bReg[4];
    auto loadRegs = [&](int c) {
        const int r  = c >> 2;                      // tap 0..8
        const int i0 = (c & 3) << 4;                // channel chunk base
        const int r3 = r / 3;
        const int dh = r3 - 1;
        const int dw = (r - r3 * 3) - 1;

        // ---- W chunk: uniform offset folded into the scalar base (GVS addressing)
        const float* __restrict__ pW = Wt + (i0 * 9 + r);
        #pragma unroll
        for (int j = 0; j < 8; ++j)
            aReg[j] = pW[aOffG[j]];

        // ---- x patch: uniform (i0, dh, dw) folded into scalar base; only the
        //      zero-padding predicate needs per-lane VALU.
        const float* __restrict__ pX = x + (i0 * NPIX + dh * WWID + dw);
        #pragma unroll
        for (int j = 0; j < 4; ++j) {
            float v = 0.0f;
            if ((unsigned)(hB[j] + dh) < (unsigned)HH &&
                (unsigned)(wB[j] + dw) < (unsigned)WWID)
                v = pX[xOffV[j]];
            bReg[j] = v;
        }
    };

    v8f acc0 = {}, acc1 = {}, acc2 = {}, acc3 = {};

    const int aFragBase = (obase + l15) * APITCH + kqoff;   // + ks*4
    const int bFragBase = lhalf * BPITCH + l15 * 2;         // + ks*2*BPITCH + t*32

    loadRegs(0);
    for (int c = 0; c < NCHUNK; ++c) {
        __syncthreads();                 // previous chunk fully consumed
        #pragma unroll
        for (int j = 0; j < 8; ++j) ALds[aOffL[j]] = aReg[j];
        #pragma unroll
        for (int j = 0; j < 4; ++j) BLds[bOffL[j]] = bReg[j];
        if (c + 1 < NCHUNK) loadRegs(c + 1);    // global loads overlap the WMMAs below
        __syncthreads();

        #pragma unroll
        for (int ks = 0; ks < 4; ++ks) {
            const v2f a  = *(const v2f*)&ALds[aFragBase + ks * 4];
            const int bb = bFragBase + ks * 2 * BPITCH;
            const v2f b0 = *(const v2f*)&BLds[bb +  0];
            const v2f b1 = *(const v2f*)&BLds[bb + 32];
            const v2f b2 = *(const v2f*)&BLds[bb + 64];
            const v2f b3 = *(const v2f*)&BLds[bb + 96];
            acc0 = __builtin_amdgcn_wmma_f32_16x16x4_f32(false, a, false, b0,
                                                         (short)0, acc0, false, false);
            acc1 = __builtin_amdgcn_wmma_f32_16x16x4_f32(false, a, false, b1,
                                                         (short)0, acc1, false, false);
            acc2 = __builtin_amdgcn_wmma_f32_16x16x4_f32(false, a, false, b2,
                                                         (short)0, acc2, false, false);
            acc3 = __builtin_amdgcn_wmma_f32_16x16x4_f32(false, a, false, b3,
                                                         (short)0, acc3, false, false);
        }
    }

    // ---- C/D layout: VGPR rr -> M = rr + 8*lhalf, N = l15 ; add bias, store NCHW
    const int orow = obase + lhalf * 8;
    #pragma unroll
    for (int rr = 0; rr < 8; ++rr) {
        int    o  = orow + rr;
        float  bv = bias[o];
        float* op = out + ((size_t)(b * OUT_CH + o) * NPIX) + n0;
        op[ 0 + l15] = acc0[rr] + bv;
        op[16 + l15] = acc1[rr] + bv;
        op[32 + l15] = acc2[rr] + bv;
        op[48 + l15] = acc3[rr] + bv;
    }
}

extern "C" void kernel_launch(void* const* d_in, const int* in_sizes, int n_in,
                              void* d_out, int out_size, void* d_ws, size_t ws_size,
                              hipStream_t stream) {
    const float* x    = (const float*)d_in[0];
    const float* Wt   = (const float*)d_in[1];
    const float* bias = (const float*)d_in[2];
    float* out        = (float*)d_out;

    dim3 grid(NPIX / NTILE, NB);   // 49 x 4 blocks
    dim3 block(256);               // 8 wave32
    conv3x3_wmma_f32<<<grid, block, 0, stream>>>(x, Wt, bias, out);
}